// QuanvolutionHybridClassifier_65481071395773
// MI455X (gfx1250) — compile-verified
//
#include <hip/hip_runtime.h>
#include <hip/hip_bf16.h>

typedef __attribute__((ext_vector_type(2))) float v2f;
typedef __attribute__((ext_vector_type(8))) float v8f;

#define GAMMA_K 1.0f

// Fused: patch-conv -> per-position RBF Gram features -> FC (f32 WMMA) -> log_softmax
// Single wave (32 lanes): problem is ~50KB total, pure latency; everything lives in LDS.
__global__ __launch_bounds__(32)
void quanv_fused_kernel(const float* __restrict__ x,       // [4,1,28,28]
                        const float* __restrict__ conv_w,  // [4,1,2,2]
                        const float* __restrict__ fc_w,    // [10,784]
                        const float* __restrict__ fc_b,    // [10]
                        float* __restrict__ out)           // [4,10] log_softmax
{
    __shared__ float s_p[196 * 16];     // p[n][b*4 + c]
    __shared__ float s_feat[4 * 784];   // feat[b][k], k = n*4 + bp
    __shared__ float s_fcw[10 * 784];   // fc_w row-major copy
    __shared__ float s_logits[4 * 16];

    const int lane = threadIdx.x;       // 0..31, wave32

    // ---- Stage fc_w into LDS (7840 floats, 245 per lane) ----
    for (int t = lane; t < 10 * 784; t += 32)
        s_fcw[t] = fc_w[t];

    // ---- Phase A: 2x2 stride-2 conv: p[b,c,i,j], stored as s_p[n*16 + b*4 + c] ----
    // 3136 values, 98 per lane; trip count uniform (no divergence).
    for (int t = lane; t < 3136; t += 32) {
        const int n  = t >> 4;          // 0..195
        const int bc = t & 15;
        const int b  = bc >> 2;
        const int c  = bc & 3;
        const int i  = n / 14;
        const int j  = n - i * 14;
        const float* xb = x + b * (28 * 28) + (2 * i) * 28 + 2 * j;
        const float* wc = conv_w + c * 4;
        s_p[t] = xb[0] * wc[0] + xb[1] * wc[1] + xb[28] * wc[2] + xb[29] * wc[3];
    }
    __syncthreads();

    // ---- Phase B: RBF Gram over batch per position ----
    // feat[b][n*4+bp] = exp(-gamma * ||p[b,n,:]-p[bp,n,:]||^2); 3136 vals, 98/lane.
    for (int t = lane; t < 3136; t += 32) {
        const int b  = t / 784;
        const int k  = t - b * 784;
        const int n  = k >> 2;
        const int bp = k & 3;
        const float* pn = s_p + n * 16;
        const float d0 = pn[b * 4 + 0] - pn[bp * 4 + 0];
        const float d1 = pn[b * 4 + 1] - pn[bp * 4 + 1];
        const float d2 = pn[b * 4 + 2] - pn[bp * 4 + 2];
        const float d3 = pn[b * 4 + 3] - pn[bp * 4 + 3];
        const float dist2 = d0 * d0 + d1 * d1 + d2 * d2 + d3 * d3;
        s_feat[t] = __expf(-GAMMA_K * dist2);
    }
    __syncthreads();

    // ---- Phase C: logits[4,10] = feat[4,784] @ fc_w^T[784,10] via f32 WMMA ----
    // One 16x16 accumulator tile, K=784 as 196 steps of V_WMMA_F32_16X16X4_F32.
    // A 16x4 layout: lane holds M=lane&15; VGPR0/1 = K={0,1}+2*(lane>=16) + kbase.
    // B 4x16 layout (lane-striped rows): N=lane&15, same K halves.
    //
    // No guards needed for the dead rows/cols: garbage in A-row m only pollutes
    // D-row m; garbage in B-col n only pollutes D-col n. We read only D[0..3][0..9],
    // so lanes with m>=4 / m>=10 just re-read a valid (duplicated) row — clamp the
    // base pointer once, keep the inner loop branch-free with b64 LDS loads.
    const int m     = lane & 15;
    const int khalf = (lane >> 4) << 1;                 // 0 or 2
    const int ma    = (m < 4)  ? m : 3;                 // clamp: dup row, dead output
    const int mb    = (m < 10) ? m : 9;
    const float* fa = s_feat + ma * 784 + khalf;
    const float* fb = s_fcw  + mb * 784 + khalf;

    v8f acc = {};
    #pragma unroll 4
    for (int kk = 0; kk < 196; ++kk) {
        const v2f a  = *(const v2f*)(fa + (kk << 2));   // ds_load_b64, 8B aligned
        const v2f bm = *(const v2f*)(fb + (kk << 2));
        acc = __builtin_amdgcn_wmma_f32_16x16x4_f32(
            /*neg_a=*/false, a, /*neg_b=*/false, bm,
            /*c_mod=*/(short)0, acc, /*reuse_a=*/false, /*reuse_b=*/false);
    }

    // ---- Phase D: spill D tile. Lanes 0-15, VGPR r holds logits[M=r][N=lane]. ----
    if (lane < 16) {
        s_logits[0 * 16 + m] = acc[0];
        s_logits[1 * 16 + m] = acc[1];
        s_logits[2 * 16 + m] = acc[2];
        s_logits[3 * 16 + m] = acc[3];
    }
    __syncthreads();

    // ---- Phase E: bias + log_softmax per row (4 rows x 10 logits) ----
    if (lane < 4) {
        float z[10];
        float mx = -3.402823466e+38f;
        #pragma unroll
        for (int nn = 0; nn < 10; ++nn) {
            z[nn] = s_logits[lane * 16 + nn] + fc_b[nn];
            mx = fmaxf(mx, z[nn]);
        }
        float se = 0.0f;
        #pragma unroll
        for (int nn = 0; nn < 10; ++nn)
            se += __expf(z[nn] - mx);
        const float lse = mx + __logf(se);
        #pragma unroll
        for (int nn = 0; nn < 10; ++nn)
            out[lane * 10 + nn] = z[nn] - lse;
    }
}

extern "C" void kernel_launch(void* const* d_in, const int* in_sizes, int n_in,
                              void* d_out, int out_size, void* d_ws, size_t ws_size,
                              hipStream_t stream) {
    (void)in_sizes; (void)n_in; (void)out_size; (void)d_ws; (void)ws_size;
    const float* x      = (const float*)d_in[0];
    const float* conv_w = (const float*)d_in[1];
    const float* fc_w   = (const float*)d_in[2];
    const float* fc_b   = (const float*)d_in[3];
    float* out = (float*)d_out;
    quanv_fused_kernel<<<1, 32, 0, stream>>>(x, conv_w, fc_w, fc_b, out);
}